// ParallelEmbedding_38225208934973
// MI455X (gfx1250) — compile-verified
//
#include <hip/hip_runtime.h>
#include <hip/hip_bf16.h>
#include <stdint.h>

typedef __attribute__((ext_vector_type(2))) float v2f;
typedef __attribute__((ext_vector_type(8))) float v8f;

#define DIM   4096
#define RANK  256
#define NROWS 8192          // BATCH * SEQ = 4 * 2048
#define ALD   260           // padded LDS row stride (mod 64 == 4 -> conflict-free b64 reads)

// out[r, :] = pad[r] ? 0 : W[x[r], :] + A[x[r], :] @ B
// Block: 16 gathered rows x 512 columns, 8 waves. Each wave runs 2 passes of a
// 16x32 tile: one K-loop (256, step 4) feeding TWO v_wmma_f32_16x16x4_f32 per
// A fragment. A rows staged to LDS with GLOBAL_LOAD_ASYNC_TO_LDS_B128
// (ASYNCcnt) via inline asm on gfx1250.
__global__ __launch_bounds__(256) void lora_embed_wmma(
    const int* __restrict__ xidx,
    const unsigned char* __restrict__ pad,
    const float* __restrict__ W,
    const float* __restrict__ A,
    const float* __restrict__ B,
    float* __restrict__ out)
{
    __shared__ float sA[16 * ALD];
    __shared__ int   sIdx[16];
    __shared__ int   sMask[16];

    const int tid     = threadIdx.x;
    const int wave    = tid >> 5;          // 0..7
    const int lane    = tid & 31;
    const int rowBase = blockIdx.y << 4;   // 16 output rows per block
    const int nSlab   = blockIdx.x << 9;   // 512 columns per block

    if (tid < 16) {
        const int r = rowBase + tid;
        sIdx[tid]  = xidx[r];
        sMask[tid] = (int)pad[r];
    }
    __syncthreads();

    // Stage the 16 gathered A rows into LDS (16B chunks, padded row stride).
    #pragma unroll
    for (int i = 0; i < 4; ++i) {
        const int lin = tid + (i << 8);    // 0..1023 float4 chunks
        const int m   = lin >> 6;          // gathered row 0..15
        const int k4  = lin & 63;          // float4 index within the 256-wide row
        const float* gsrc = A + (size_t)sIdx[m] * RANK + (k4 << 2);
        float*       ldst = &sA[m * ALD + (k4 << 2)];
#if defined(__gfx1250__)
        // Async DMA: memory -> LDS, tracked with ASYNCcnt. VDST = LDS byte
        // offset (low 32 bits of the generic shared pointer), VADDR = 64-bit
        // global address.
        const uint32_t la = (uint32_t)(uintptr_t)ldst;
        const uint64_t ga = (uint64_t)(uintptr_t)gsrc;
        asm volatile("global_load_async_to_lds_b128 %0, %1, off"
                     :
                     : "v"(la), "v"(ga)
                     : "memory");
#else
        *reinterpret_cast<float4*>(ldst) = *reinterpret_cast<const float4*>(gsrc);
#endif
    }
#if defined(__gfx1250__)
#if __has_builtin(__builtin_amdgcn_s_wait_asynccnt)
    __builtin_amdgcn_s_wait_asynccnt(0);
#else
    asm volatile("s_wait_asynccnt 0x0" ::: "memory");
#endif
#endif
    __syncthreads();

    // WMMA f32 16x16x4 fragment roles for this lane:
    //  A: row M = lane&15, k-pair base = 2*(lane>>4)   (v0=k, v1=k+1)
    //  B: col N = lane&15, same k-pair
    //  C/D: VGPR i -> M = i + 8*(lane>>4), N = lane&15
    const int nlane = lane & 15;
    const int khalf = lane >> 4;           // 0 or 1
    const int mC0   = khalf << 3;          // C-row base for this half-wave
    const float* aBase = &sA[(lane & 15) * ALD + (khalf << 1)];

    // Hoist gathered weight-row offsets (shared by both passes).
    size_t wRow[8];
    #pragma unroll
    for (int i = 0; i < 8; ++i)
        wRow[i] = (size_t)sIdx[i + mC0] * DIM;

    #pragma unroll
    for (int t = 0; t < 2; ++t) {
        const int n = nSlab + (wave << 5) + (t << 8) + nlane;   // left tile column

        // Seed both accumulators with the gathered weight tiles (C layout).
        v8f c0, c1;
        #pragma unroll
        for (int i = 0; i < 8; ++i) {
            c0[i] = W[wRow[i] + n];
            c1[i] = W[wRow[i] + n + 16];
        }

        // Rank-256 correction: one A fragment drives two WMMAs (cols n, n+16).
        const float* bPtr = B + (size_t)(khalf << 1) * DIM + n;
        #pragma unroll 8
        for (int kk = 0; kk < RANK; kk += 4) {
            const float2 af = *reinterpret_cast<const float2*>(aBase + kk);
            v2f a; a.x = af.x; a.y = af.y;
            const float* p0 = bPtr + (size_t)kk * DIM;
            v2f b0; b0.x = p0[0];  b0.y = p0[DIM];
            v2f b1; b1.x = p0[16]; b1.y = p0[DIM + 16];
            c0 = __builtin_amdgcn_wmma_f32_16x16x4_f32(
                     false, a, false, b0, (short)0, c0, false, false);
            c1 = __builtin_amdgcn_wmma_f32_16x16x4_f32(
                     false, a, false, b1, (short)0, c1, false, false);
        }

        // Store with padding mask applied per row (output rows contiguous).
        float* oBase = out + (size_t)rowBase * DIM + n;
        #pragma unroll
        for (int i = 0; i < 8; ++i) {
            const int m = i + mC0;
            oBase[(size_t)m * DIM]      = sMask[m] ? 0.0f : c0[i];
            oBase[(size_t)m * DIM + 16] = sMask[m] ? 0.0f : c1[i];
        }
    }
}

extern "C" void kernel_launch(void* const* d_in, const int* in_sizes, int n_in,
                              void* d_out, int out_size, void* d_ws, size_t ws_size,
                              hipStream_t stream) {
    const int*           x    = (const int*)d_in[0];
    const unsigned char* padm = (const unsigned char*)d_in[1];  // jax bool = 1 byte
    const float*         W    = (const float*)d_in[2];
    const float*         A    = (const float*)d_in[3];
    const float*         B    = (const float*)d_in[4];
    float*               out  = (float*)d_out;

    dim3 grid(DIM / 512, NROWS / 16);   // (8, 512)
    lora_embed_wmma<<<grid, 256, 0, stream>>>(x, padm, W, A, B, out);
}